// ThreeEDisInit_70334384439574
// MI455X (gfx1250) — compile-verified
//
#include <hip/hip_runtime.h>

typedef __attribute__((ext_vector_type(2))) float v2f;
typedef __attribute__((ext_vector_type(8))) float v8f;

#define B_DIM 2
#define N_DIM 64
#define Z_DIM 128
#define E_DIM 64
#define K_DIM 64
#define LDST 68   // padded LDS row stride (floats): conflict-light, keeps 8B alignment

#define BNK  (B_DIM * N_DIM * K_DIM)           // 8192
#define BNNK (B_DIM * N_DIM * N_DIM * K_DIM)   // 524288

__device__ __forceinline__ float silu_f(float x) {
    return x / (1.0f + __expf(-x));
}

// ---------------------------------------------------------------------------
// Stage 1: z0/z1/z2 = silu(emb1 @ Wz[lin] + bz[lin]).  Tiny (24576 outputs).
// z_ws layout: [lin][b][n][k]
// ---------------------------------------------------------------------------
__global__ void ThreeEDis_prep_z(const float* __restrict__ emb1,
                                 const float* __restrict__ Wz,
                                 const float* __restrict__ bz,
                                 float* __restrict__ z_ws) {
    int idx = blockIdx.x * blockDim.x + threadIdx.x;
    if (idx >= 3 * B_DIM * N_DIM * K_DIM) return;
    int k   = idx & 63;
    int t   = idx >> 6;
    int n   = t & 63;
    int t2  = t >> 6;
    int b   = t2 & 1;
    int lin = t2 >> 1;
    const float* er = emb1 + (b * N_DIM + n) * Z_DIM;
    const float* wc = Wz + lin * Z_DIM * K_DIM + k;
    float acc = bz[lin * K_DIM + k];
#pragma unroll 4
    for (int zz = 0; zz < Z_DIM; ++zz) acc += er[zz] * wc[zz * K_DIM];
    z_ws[idx] = silu_f(acc);
}

// ---------------------------------------------------------------------------
// Stage 2: e[lin] = ef @ Wef[lin]  (no bias, no act).  M=8192 rows, K=64, N=64.
// WMMA f32 16x16x4. Each wave: one 16-row x 64-col tile. 4 waves / block.
// e_ws layout: [lin][b][n1][n2][k]
// ---------------------------------------------------------------------------
__global__ __launch_bounds__(128)
void ThreeEDis_prep_e(const float* __restrict__ ef,
                      const float* __restrict__ Wef,
                      float* __restrict__ e_ws) {
    __shared__ float wT[K_DIM * LDST];   // Wef[lin] transposed: wT[n][e]
    const int lin = blockIdx.y;
    const int tid = threadIdx.x;
    for (int t = tid; t < E_DIM * K_DIM; t += blockDim.x) {
        int e = t >> 6, n = t & 63;
        wT[n * LDST + e] = Wef[lin * E_DIM * K_DIM + e * K_DIM + n];
    }
    __syncthreads();

    const int wave = tid >> 5, lane = tid & 31;
    const int m = lane & 15, half = lane >> 4;
    const int r0 = blockIdx.x * 64 + wave * 16;        // 16-row tile base
    const float* arow = ef + (size_t)(r0 + m) * E_DIM; // rows contiguous in E

    v2f a[16];
#pragma unroll
    for (int c = 0; c < 16; ++c)
        a[c] = *(const v2f*)(arow + 4 * c + 2 * half);

    float* outp = e_ws + (size_t)lin * BNNK;
    const int nl = lane & 15;
#pragma unroll
    for (int dn = 0; dn < 4; ++dn) {
        const int n = 16 * dn + nl;
        v8f acc = {0.f,0.f,0.f,0.f,0.f,0.f,0.f,0.f};
#pragma unroll
        for (int c = 0; c < 16; ++c) {
            v2f bf = *(const v2f*)(&wT[n * LDST + 4 * c + 2 * half]);
            acc = __builtin_amdgcn_wmma_f32_16x16x4_f32(
                false, a[c], false, bf, (short)0, acc, false, false);
        }
#pragma unroll
        for (int v = 0; v < 8; ++v) {
            int orow = r0 + v + 8 * half;  // C/D layout: vgpr v -> M=v (lanes<16) / v+8
            outp[(size_t)orow * K_DIM + n] = acc[v];
        }
    }
}

// ---------------------------------------------------------------------------
// Stage 3 (fused, dominant): build x on the fly, 2-layer SiLU MLP via WMMA f32,
// residual add, store. Block = 64 threads (2 waves) handling 32 rows (fixed
// b,i,j, half of the l-range). Grid = 16384 blocks.
// ---------------------------------------------------------------------------
__global__ __launch_bounds__(64)
void ThreeEDis_fused_main(const float* __restrict__ z_ws,
                          const float* __restrict__ e_ws,
                          const float* __restrict__ table,
                          const float* __restrict__ Wm,
                          const float* __restrict__ bm,
                          float* __restrict__ out) {
    __shared__ float wT0[K_DIM * LDST];      // Wm[0]^T : wT0[n][kd]
    __shared__ float wT1[K_DIM * LDST];      // Wm[1]^T
    __shared__ float wij[K_DIM];             // z1[b,i,k]*z2[b,j,k]*e2[b,i,j,k]
    __shared__ float xls[2][16 * LDST];      // per-wave x tile (residual relayout)
    __shared__ float hls[2][16 * LDST];      // per-wave h tile (layer-1 relayout)

    const int tid = threadIdx.x;
    const int bid = blockIdx.x;
    const int lh = bid & 1;
    const int j  = (bid >> 1) & 63;
    const int i  = (bid >> 7) & 63;
    const int b  = bid >> 13;

    for (int t = tid; t < K_DIM * K_DIM; t += blockDim.x) {
        int kd = t >> 6, n = t & 63;
        wT0[n * LDST + kd] = Wm[kd * K_DIM + n];
        wT1[n * LDST + kd] = Wm[K_DIM * K_DIM + kd * K_DIM + n];
    }
    if (tid < K_DIM) {
        int k = tid;
        float z1v = z_ws[BNK     + (b * N_DIM + i) * K_DIM + k];
        float z2v = z_ws[2 * BNK + (b * N_DIM + j) * K_DIM + k];
        float e2v = e_ws[2 * (size_t)BNNK + ((size_t)((b * N_DIM + i) * N_DIM + j)) * K_DIM + k];
        wij[k] = z1v * z2v * e2v;
    }
    __syncthreads();

    const int wave = tid >> 5, lane = tid & 31;
    const int m = lane & 15, half = lane >> 4;
    const int l = 32 * lh + 16 * wave + m;     // third spatial index of this row

    // pattern id with the reference's "later write wins" semantics
    int p = 1;
    if (i == j && j == l) p = 2;
    if (i == l)           p = 3;
    if (j == l)           p = 4;
    if (i == j)           p = 5;
    const float* tab = table + p * K_DIM;

    const float* z0r = z_ws + (b * N_DIM + l) * K_DIM;
    const float* e0r = e_ws + ((size_t)((b * N_DIM + j) * N_DIM + l)) * K_DIM;
    const float* e1r = e_ws + (size_t)BNNK + ((size_t)((b * N_DIM + i) * N_DIM + l)) * K_DIM;

    float* xw = &xls[wave][0];
    float* hw = &hls[wave][0];

    // ---- build x in WMMA A-layout (registers) + mirror to LDS for residual ----
    v2f a[16];
#pragma unroll
    for (int c = 0; c < 16; ++c) {
        int k0 = 4 * c + 2 * half;
        v2f z0v = *(const v2f*)(z0r + k0);
        v2f e0v = *(const v2f*)(e0r + k0);
        v2f e1v = *(const v2f*)(e1r + k0);
        v2f tv  = *(const v2f*)(tab + k0);
        v2f wv  = *(const v2f*)(&wij[k0]);
        v2f xv;
        xv.x = wv.x * z0v.x * e0v.x * e1v.x * tv.x;
        xv.y = wv.y * z0v.y * e0v.y * e1v.y * tv.y;
        *(v2f*)(&xw[m * LDST + k0]) = xv;
        a[c] = xv;
    }

    const int nl = lane & 15;

    // ---- layer 1: h = silu(x @ Wm0 + bm0), written to LDS in D-layout ----
#pragma unroll
    for (int dn = 0; dn < 4; ++dn) {
        const int n = 16 * dn + nl;
        v8f acc = {0.f,0.f,0.f,0.f,0.f,0.f,0.f,0.f};
#pragma unroll
        for (int c = 0; c < 16; ++c) {
            v2f bf = *(const v2f*)(&wT0[n * LDST + 4 * c + 2 * half]);
            acc = __builtin_amdgcn_wmma_f32_16x16x4_f32(
                false, a[c], false, bf, (short)0, acc, false, false);
        }
        float bb = bm[n];
#pragma unroll
        for (int v = 0; v < 8; ++v)
            hw[(v + 8 * half) * LDST + n] = silu_f(acc[v] + bb);
    }
    __syncthreads();

    // ---- reload h as A-fragments ----
#pragma unroll
    for (int c = 0; c < 16; ++c)
        a[c] = *(const v2f*)(&hw[m * LDST + 4 * c + 2 * half]);

    // ---- layer 2 + residual + store ----
    float* ob = out + ((size_t)((b * N_DIM + i) * N_DIM + j) * N_DIM + 32 * lh + 16 * wave) * K_DIM;
#pragma unroll
    for (int dn = 0; dn < 4; ++dn) {
        const int n = 16 * dn + nl;
        v8f acc = {0.f,0.f,0.f,0.f,0.f,0.f,0.f,0.f};
#pragma unroll
        for (int c = 0; c < 16; ++c) {
            v2f bf = *(const v2f*)(&wT1[n * LDST + 4 * c + 2 * half]);
            acc = __builtin_amdgcn_wmma_f32_16x16x4_f32(
                false, a[c], false, bf, (short)0, acc, false, false);
        }
        float bb = bm[K_DIM + n];
#pragma unroll
        for (int v = 0; v < 8; ++v) {
            int r = v + 8 * half;
            float hv = silu_f(acc[v] + bb);
            ob[(size_t)r * K_DIM + n] = hv + xw[r * LDST + n];
        }
    }
}

// ---------------------------------------------------------------------------
extern "C" void kernel_launch(void* const* d_in, const int* in_sizes, int n_in,
                              void* d_out, int out_size, void* d_ws, size_t ws_size,
                              hipStream_t stream) {
    const float* emb1  = (const float*)d_in[0];
    const float* ef    = (const float*)d_in[1];
    const float* Wz    = (const float*)d_in[2];
    const float* bz    = (const float*)d_in[3];
    const float* Wef   = (const float*)d_in[4];
    const float* table = (const float*)d_in[5];
    const float* Wm    = (const float*)d_in[6];
    const float* bm    = (const float*)d_in[7];
    float* out = (float*)d_out;

    float* z_ws = (float*)d_ws;              // 3 * 8192 floats
    float* e_ws = z_ws + 3 * BNK;            // 3 * 524288 floats (~6 MB, L2-resident)

    ThreeEDis_prep_z<<<(3 * B_DIM * N_DIM * K_DIM + 255) / 256, 256, 0, stream>>>(
        emb1, Wz, bz, z_ws);
    ThreeEDis_prep_e<<<dim3(128, 3), 128, 0, stream>>>(ef, Wef, e_ws);
    ThreeEDis_fused_main<<<B_DIM * N_DIM * N_DIM * 2, 64, 0, stream>>>(
        z_ws, e_ws, table, Wm, bm, out);
}